// SfMPoseLoss_76879914599137
// MI455X (gfx1250) — compile-verified
//
#include <hip/hip_runtime.h>
#include <math.h>

typedef float v2f __attribute__((ext_vector_type(2)));
typedef float v8f __attribute__((ext_vector_type(8)));

#define WPB  8    // waves per block in pass 1 (1 wave == 1 segment)
#define LSTR 17   // LDS float stride per staged point (16 comps + 1 pad, conflict-free)

// ---------------- Kernel 1: exclusive scan of counts -> starts ----------------
__global__ void k_scan_starts(const int* __restrict__ counts, int* __restrict__ starts, int E) {
  __shared__ int sh[1024];
  const int t = threadIdx.x;
  int carry = 0;
  for (int base = 0; base < E; base += 1024) {
    const int v = (base + t < E) ? counts[base + t] : 0;
    sh[t] = v;
    __syncthreads();
    for (int off = 1; off < 1024; off <<= 1) {
      const int add = (t >= off) ? sh[t - off] : 0;
      __syncthreads();
      sh[t] += add;
      __syncthreads();
    }
    if (base + t < E) starts[base + t] = carry + sh[t] - v;
    carry += sh[1023];
    __syncthreads();
  }
}

// rotate up=(0,1,0) by conj(q)  (matches reference q_rot(q_inv(q), UP))
__device__ inline void qrot_conj_up(float qx, float qy, float qz, float qw, float o[3]) {
  const float vx = -qx, vy = -qy, vz = -qz;
  // t = 2 * cross(qv, up), up = (0,1,0)
  const float tx = 2.f * (-vz);
  const float ty = 0.f;
  const float tz = 2.f * ( vx);
  o[0] = 0.f + qw * tx + (vy * tz - vz * ty);
  o[1] = 1.f + qw * ty + (vz * tx - vx * tz);
  o[2] = 0.f + qw * tz + (vx * ty - vy * tx);
}

// ---------------- Kernel 2: per-segment G = sum(u u^T) via WMMA f32 16x16x4 ----------------
__global__ void __launch_bounds__(256) k_pass1(
    const float* __restrict__ pred_t, const float* __restrict__ pred_q,
    const float* __restrict__ gt_t,   const float* __restrict__ gt_q,
    const int* __restrict__ counts,   const int* __restrict__ starts,
    float* __restrict__ gbuf, int E)
{
  __shared__ float lds[WPB * 32 * LSTR];
  const int lane = threadIdx.x & 31;
  const int wave = threadIdx.x >> 5;
  const int e = blockIdx.x * WPB + wave;
  if (e >= E) return;                    // wave-uniform exit; EXEC stays all-ones inside

  float* st = &lds[wave * 32 * LSTR];
  const int n    = counts[e];
  const int base = starts[e];
  const int c = lane & 15;               // matrix row/col component this lane feeds
  const int h = lane >> 4;               // half-wave select (K pairing per ISA A layout)

  // padding components u[12..15] are constant zero: stage them once, outside the loop
  st[lane * LSTR + 12] = 0.f;
  st[lane * LSTR + 13] = 0.f;
  st[lane * LSTR + 14] = 0.f;
  st[lane * LSTR + 15] = 0.f;

  v8f acc = {0.f, 0.f, 0.f, 0.f, 0.f, 0.f, 0.f, 0.f};

  const int tiles = (n + 31) >> 5;
  for (int t = 0; t < tiles; ++t) {
    const int li = t * 32 + lane;
    const bool active = li < n;

    if (t + 1 < tiles) {                 // speculative prefetch of next tile (global_prefetch_b8)
      const int ni = base + (t + 1) * 32 + lane;
      __builtin_prefetch(&pred_q[4 * ni], 0, 1);
      __builtin_prefetch(&gt_q[4 * ni], 0, 1);
    }

    float u[12];
#pragma unroll
    for (int j = 0; j < 12; ++j) u[j] = 0.f;

    if (active) {
      const int i = base + li;
      const float px = pred_t[3 * i + 0], py = pred_t[3 * i + 1], pz = pred_t[3 * i + 2];
      const float gx = gt_t[3 * i + 0],  gy = gt_t[3 * i + 1],  gz = gt_t[3 * i + 2];
      const float4 P = ((const float4*)pred_q)[i];
      const float4 G = ((const float4*)gt_q)[i];
      // v_k = (e_k (x) pred_q) . gt_q   (bilinear form of the rot loss)
      const float vx =  P.w * G.x - P.z * G.y + P.y * G.z - P.x * G.w;
      const float vy =  P.z * G.x + P.w * G.y - P.x * G.z - P.y * G.w;
      const float vz = -P.y * G.x + P.x * G.y + P.w * G.z - P.z * G.w;
      const float vw =  P.x * G.x + P.y * G.y + P.z * G.z + P.w * G.w;
      // gravity term
      float pu[3], gu[3];
      qrot_conj_up(P.x, P.y, P.z, P.w, pu);
      qrot_conj_up(G.x, G.y, G.z, G.w, gu);
      const float np = sqrtf(pu[0]*pu[0] + pu[1]*pu[1] + pu[2]*pu[2]);
      const float ng = sqrtf(gu[0]*gu[0] + gu[1]*gu[1] + gu[2]*gu[2]);
      const float cosv = (pu[0]*gu[0] + pu[1]*gu[1] + pu[2]*gu[2]) / fmaxf(np * ng, 1e-8f);
      u[0] = px; u[1] = py; u[2] = pz;
      u[3] = gx; u[4] = gy; u[5] = gz;
      u[6] = vx; u[7] = vy; u[8] = vz; u[9] = vw;
      u[10] = 1.f;
      u[11] = 1.f - cosv;
    }

#pragma unroll
    for (int j = 0; j < 12; ++j) st[lane * LSTR + j] = u[j];
    asm volatile("s_wait_dscnt 0x0" ::: "memory");   // stores visible before transposed reads

#pragma unroll
    for (int g = 0; g < 8; ++g) {
      // ISA f32 A(16x4) layout: v0 = K{0|2}, v1 = K{1|3} split across lane halves.
      const int p0 = 4 * g + 2 * h;
      v2f a;
      a[0] = st[(p0)     * LSTR + c];
      a[1] = st[(p0 + 1) * LSTR + c];
      // G += A * A^T : B register image of U^T is identical to A's image of U.
      acc = __builtin_amdgcn_wmma_f32_16x16x4_f32(false, a, false, a, (short)0, acc, false, false);
    }
    asm volatile("s_wait_dscnt 0x0" ::: "memory");   // reads done before next tile's writes
  }

  float* out = gbuf + (size_t)e * 256;
#pragma unroll
  for (int r = 0; r < 8; ++r) out[r * 32 + lane] = acc[r];  // C/D layout: vgpr r = rows {r, r+8}
}

// ---------------- Kernel 3: per-segment finalize (Davenport-q replaces 3x3 SVD) ----------------
__global__ void k_finalize(
    const float* __restrict__ pred_t, const float* __restrict__ pred_q,
    const float* __restrict__ gt_q,
    const int* __restrict__ counts, const int* __restrict__ starts,
    const float* __restrict__ gbuf, float* __restrict__ segloss, int E)
{
  const int e = blockIdx.x * blockDim.x + threadIdx.x;
  if (e >= E) return;
  const float* g = gbuf + (size_t)e * 256;
  auto GM = [&](int m, int nn) -> float {
    const int r = m & 7;
    const int col = (m < 8) ? nn : nn + 16;
    return g[r * 32 + col];
  };
  const int   ni = counts[e];
  const float nf = (float)ni;
  const int   s0 = starts[e];

  const float sp0 = GM(0, 10), sp1 = GM(1, 10), sp2 = GM(2, 10);
  const float sg0 = GM(3, 10), sg1 = GM(4, 10), sg2 = GM(5, 10);
  const float spp = GM(0, 0) + GM(1, 1) + GM(2, 2);
  const float sgg = GM(3, 3) + GM(4, 4) + GM(5, 5);
  float H[3][3];
  const float spv[3] = {sp0, sp1, sp2}, sgv[3] = {sg0, sg1, sg2};
#pragma unroll
  for (int i = 0; i < 3; ++i)
#pragma unroll
    for (int j = 0; j < 3; ++j)
      H[i][j] = GM(i, 3 + j) - spv[i] * sgv[j] / nf;

  const float denom = spp - (sp0 * sp0 + sp1 * sp1 + sp2 * sp2) / nf;
  const float gss   = sgg - (sg0 * sg0 + sg1 * sg1 + sg2 * sg2) / nf;
  const float trH   = H[0][0] + H[1][1] + H[2][2];
  const float gvar  = gss / (nf - 1.f);
  const bool  deg   = (ni < 4) || (gvar < 1e-4f);

  float M44[4][4];
#pragma unroll
  for (int a = 0; a < 4; ++a)
#pragma unroll
    for (int b = 0; b < 4; ++b) M44[a][b] = GM(6 + a, 6 + b);
  const float gravsum = GM(10, 11);

  float nom = 0.f, qx = 0.f, qy = 0.f, qz = 0.f, qw = 1.f;
  if (!deg) {
    const float Sxx = H[0][0], Sxy = H[0][1], Sxz = H[0][2];
    const float Syx = H[1][0], Syy = H[1][1], Syz = H[1][2];
    const float Szx = H[2][0], Szy = H[2][1], Szz = H[2][2];
    const float N00 =  Sxx + Syy + Szz, N01 = Syz - Szy, N02 = Szx - Sxz, N03 = Sxy - Syx;
    const float N11 =  Sxx - Syy - Szz, N12 = Sxy + Syx, N13 = Szx + Sxz;
    const float N22 = -Sxx + Syy - Szz, N23 = Syz + Szy;
    const float N33 = -Sxx - Syy + Szz;
    const float fro = sqrtf(N00*N00 + N11*N11 + N22*N22 + N33*N33 +
                            2.f*(N01*N01 + N02*N02 + N03*N03 + N12*N12 + N13*N13 + N23*N23));
    const float shf = fro + 1e-20f;   // shift: makes lambda_max + shf strictly dominant
    float x0 = 0.82f, x1 = 0.31f, x2 = 0.47f, x3 = 0.11f;
    for (int it = 0; it < 48; ++it) {
      const float y0 = (N00 + shf) * x0 + N01 * x1 + N02 * x2 + N03 * x3;
      const float y1 = N01 * x0 + (N11 + shf) * x1 + N12 * x2 + N13 * x3;
      const float y2 = N02 * x0 + N12 * x1 + (N22 + shf) * x2 + N23 * x3;
      const float y3 = N03 * x0 + N13 * x1 + N23 * x2 + (N33 + shf) * x3;
      const float inv = 1.f / sqrtf(y0*y0 + y1*y1 + y2*y2 + y3*y3 + 1e-30f);
      x0 = y0 * inv; x1 = y1 * inv; x2 = y2 * inv; x3 = y3 * inv;
    }
    // Rayleigh quotient on unshifted N == nom == tr(R H_c)
    nom = x0 * (N00*x0 + N01*x1 + N02*x2 + N03*x3)
        + x1 * (N01*x0 + N11*x1 + N12*x2 + N13*x3)
        + x2 * (N02*x0 + N12*x1 + N22*x2 + N23*x3)
        + x3 * (N03*x0 + N13*x1 + N23*x2 + N33*x3);
    qw = x0; qx = x1; qy = x2; qz = x3;   // (w,x,y,z) -> reorder below
  }

  float s = 1.f;
  if (denom > 1e-6f) s = fminf(fmaxf(nom / fmaxf(denom, 1e-6f), 1e-3f), 1e3f);

  float trans = deg ? (denom - 2.f * trH + gss) : (s * s * denom - 2.f * s * nom + gss);
  trans /= (nf * 3.f);
  const float lg = logf(fabsf(s) + 1e-6f);
  const float scale = deg ? 0.f : lg * lg;

  float qe[4];
  if (deg) {
    const float4 P0 = ((const float4*)pred_q)[s0];
    const float4 G0 = ((const float4*)gt_q)[s0];
    // r = q0p (x) conj(q0g); q_eff = conj(r)
    const float cx = -G0.x, cy = -G0.y, cz = -G0.z, cw = G0.w;
    const float rx = P0.w*cx + P0.x*cw + P0.y*cz - P0.z*cy;
    const float ry = P0.w*cy - P0.x*cz + P0.y*cw + P0.z*cx;
    const float rz = P0.w*cz + P0.x*cy - P0.y*cx + P0.z*cw;
    const float rw = P0.w*cw - P0.x*cx - P0.y*cy - P0.z*cz;
    qe[0] = -rx; qe[1] = -ry; qe[2] = -rz; qe[3] = rw;
  } else {
    qe[0] = qx; qe[1] = qy; qe[2] = qz; qe[3] = qw;
  }
  float sdot2 = 0.f;
#pragma unroll
  for (int a = 0; a < 4; ++a)
#pragma unroll
    for (int b = 0; b < 4; ++b) sdot2 += qe[a] * M44[a][b] * qe[b];

  const float rot  = (nf - sdot2) / nf;
  const float grav = gravsum / nf;

  const float tx = pred_t[3 * s0 + 0], ty = pred_t[3 * s0 + 1], tz = pred_t[3 * s0 + 2];
  const float mag = sqrtf(tx * tx + ty * ty + tz * tz);
  const float hin = fmaxf(mag - 15.0f, 0.f);
  const float leash = hin * hin;

  float* o = segloss + (size_t)e * 5;
  o[0] = trans; o[1] = rot; o[2] = grav; o[3] = leash; o[4] = scale;
}

// ---------------- Kernel 4: deterministic 5-way mean over E segments ----------------
__global__ void k_reduce(const float* __restrict__ segloss, float* __restrict__ out, int E) {
  __shared__ float sh[5][256];
  const int t = threadIdx.x;
  float a[5] = {0.f, 0.f, 0.f, 0.f, 0.f};
  for (int e = t; e < E; e += 256)
#pragma unroll
    for (int j = 0; j < 5; ++j) a[j] += segloss[(size_t)e * 5 + j];
#pragma unroll
  for (int j = 0; j < 5; ++j) sh[j][t] = a[j];
  __syncthreads();
  for (int off = 128; off > 0; off >>= 1) {
    if (t < off)
#pragma unroll
      for (int j = 0; j < 5; ++j) sh[j][t] += sh[j][t + off];
    __syncthreads();
  }
  if (t < 5) out[t] = sh[t][0] / (float)E;
}

extern "C" void kernel_launch(void* const* d_in, const int* in_sizes, int n_in,
                              void* d_out, int out_size, void* d_ws, size_t ws_size,
                              hipStream_t stream) {
  (void)n_in; (void)out_size; (void)ws_size;
  const float* pred_t = (const float*)d_in[0];
  const float* pred_q = (const float*)d_in[1];
  const float* gt_t   = (const float*)d_in[2];
  const float* gt_q   = (const float*)d_in[3];
  const int*   counts = (const int*)d_in[4];
  const int E = in_sizes[4];

  char* ws = (char*)d_ws;
  int*   starts  = (int*)ws;
  size_t off     = (((size_t)E * sizeof(int)) + 255) & ~(size_t)255;
  float* gbuf    = (float*)(ws + off);
  float* segloss = gbuf + (size_t)E * 256;

  k_scan_starts<<<1, 1024, 0, stream>>>(counts, starts, E);
  const int blocks1 = (E + WPB - 1) / WPB;
  k_pass1<<<blocks1, 256, 0, stream>>>(pred_t, pred_q, gt_t, gt_q, counts, starts, gbuf, E);
  k_finalize<<<(E + 255) / 256, 256, 0, stream>>>(pred_t, pred_q, gt_q, counts, starts, gbuf, segloss, E);
  k_reduce<<<1, 256, 0, stream>>>(segloss, (float*)d_out, E);
}